// LearnedQuantizer_69758858822320
// MI455X (gfx1250) — compile-verified
//
#include <hip/hip_runtime.h>
#include <math.h>

typedef __attribute__((ext_vector_type(4))) float v4f;

#define DEV_INLINE __device__ __forceinline__

constexpr int   kD   = 128;   // head dim (reference constant)
constexpr float kEps = 1e-8f;

// Branchless nearest-centroid for 8 sorted centroids.
// Strict '>' on midpoints reproduces argmin's tie-break to the lower index.
DEV_INLINE float quant8(float v, const float (&cen)[8], const float (&bnd)[7]) {
    float q = cen[0];
#pragma unroll
    for (int i = 0; i < 7; ++i) {
        q = (v > bnd[i]) ? cen[i + 1] : q;
    }
    return q;
}

// Wave32 all-reduce sum via ds_swizzle XOR butterfly (group-of-32 mode:
// offset = xor_mask<<10 | and_mask 0x1f). One DS op per step, no index VGPRs.
DEV_INLINE float wave32_allreduce_sum(float v) {
    v += __int_as_float(__builtin_amdgcn_ds_swizzle(__float_as_int(v), 0x041F)); // xor 1
    v += __int_as_float(__builtin_amdgcn_ds_swizzle(__float_as_int(v), 0x081F)); // xor 2
    v += __int_as_float(__builtin_amdgcn_ds_swizzle(__float_as_int(v), 0x101F)); // xor 4
    v += __int_as_float(__builtin_amdgcn_ds_swizzle(__float_as_int(v), 0x201F)); // xor 8
    v += __int_as_float(__builtin_amdgcn_ds_swizzle(__float_as_int(v), 0x401F)); // xor 16
    return v;
}

__global__ __launch_bounds__(256) void learned_quantizer_fwd(
    const float* __restrict__ x,
    const float* __restrict__ angles,        // (D/2,)
    const float* __restrict__ centroids,     // (8,) sorted
    const float* __restrict__ mean_corr,     // (D,)
    const float* __restrict__ running_mean,  // (D,)
    float* __restrict__ out,
    int n_rows)
{
    const int lane          = threadIdx.x & 31;
    const int waveInBlock   = threadIdx.x >> 5;
    const int wavesPerBlock = blockDim.x >> 5;
    const int waveId        = blockIdx.x * wavesPerBlock + waveInBlock;
    const int numWaves      = gridDim.x * wavesPerBlock;

    const int col = lane << 2;  // 4 consecutive floats per lane -> 32 lanes cover D=128

    // ---- per-wave constants, hoisted out of the streaming loop ----
    const float m0 = running_mean[col + 0] + mean_corr[col + 0];
    const float m1 = running_mean[col + 1] + mean_corr[col + 1];
    const float m2 = running_mean[col + 2] + mean_corr[col + 2];
    const float m3 = running_mean[col + 3] + mean_corr[col + 3];

    // two Givens pairs per lane: pair indices 2*lane, 2*lane+1
    const float a0 = angles[(col >> 1) + 0];
    const float a1 = angles[(col >> 1) + 1];
    const float c0 = __cosf(a0), s0 = __sinf(a0);
    const float c1 = __cosf(a1), s1 = __sinf(a1);

    // centroids are wave-uniform -> scalar loads into registers
    float cen[8];
#pragma unroll
    for (int i = 0; i < 8; ++i) cen[i] = centroids[i];
    float bnd[7];
#pragma unroll
    for (int i = 0; i < 7; ++i) bnd[i] = 0.5f * (cen[i] + cen[i + 1]);

    // Process one 512B row: NT b128 in -> center/normalize/rotate/quantize/
    // unrotate/rescale -> NT b128 out.
    auto process_row = [&](int row) {
        const size_t base = (size_t)row * kD;

        const v4f v = __builtin_nontemporal_load((const v4f*)(x + base) + lane);

        const float x0 = v.x - m0;
        const float x1 = v.y - m1;
        const float x2 = v.z - m2;
        const float x3 = v.w - m3;

        const float ss   = wave32_allreduce_sum(x0 * x0 + x1 * x1 + x2 * x2 + x3 * x3);
        const float norm = fmaxf(__builtin_amdgcn_sqrtf(ss), kEps);
        const float inv  = __builtin_amdgcn_rcpf(norm);

        const float u0 = x0 * inv, u1 = x1 * inv;
        const float u2 = x2 * inv, u3 = x3 * inv;

        // forward Givens rotation (lane-local: pairs never cross the float4)
        const float r0 = c0 * u0 - s0 * u1;
        const float r1 = s0 * u0 + c0 * u1;
        const float r2 = c1 * u2 - s1 * u3;
        const float r3 = s1 * u2 + c1 * u3;

        // hard quantize (forward of straight-through estimator)
        const float q0 = quant8(r0, cen, bnd);
        const float q1 = quant8(r1, cen, bnd);
        const float q2 = quant8(r2, cen, bnd);
        const float q3 = quant8(r3, cen, bnd);

        // inverse rotation (s -> -s)
        const float y0 =  c0 * q0 + s0 * q1;
        const float y1 = -s0 * q0 + c0 * q1;
        const float y2 =  c1 * q2 + s1 * q3;
        const float y3 = -s1 * q2 + c1 * q3;

        v4f o;
        o.x = fmaf(y0, norm, m0);
        o.y = fmaf(y1, norm, m1);
        o.z = fmaf(y2, norm, m2);
        o.w = fmaf(y3, norm, m3);

        __builtin_nontemporal_store(o, (v4f*)(out + base) + lane);
    };

    // ---- streaming loop, unrolled x2 for memory-level parallelism ----
    int row = waveId;
    for (; row + numWaves < n_rows; row += 2 * numWaves) {
        // prefetch the next unrolled pair (global_prefetch_b8)
        const int pf = row + 2 * numWaves;
        if (pf < n_rows) {
            __builtin_prefetch(x + (size_t)pf * kD + col, 0, 0);
        }
        const int pf2 = pf + numWaves;
        if (pf2 < n_rows) {
            __builtin_prefetch(x + (size_t)pf2 * kD + col, 0, 0);
        }
        process_row(row);
        process_row(row + numWaves);
    }
    if (row < n_rows) {
        process_row(row);
    }
}

extern "C" void kernel_launch(void* const* d_in, const int* in_sizes, int n_in,
                              void* d_out, int out_size, void* d_ws, size_t ws_size,
                              hipStream_t stream) {
    const float* x            = (const float*)d_in[0];
    const float* angles       = (const float*)d_in[1];
    const float* centroids    = (const float*)d_in[2];
    const float* mean_corr    = (const float*)d_in[3];
    const float* running_mean = (const float*)d_in[4];
    float*       out          = (float*)d_out;

    const int halfD  = in_sizes[1];          // D/2 rotation angles
    const int Ddim   = 2 * halfD;            // == 128 for this problem
    const int n_rows = in_sizes[0] / Ddim;   // == 262144

    const int block         = 256;           // 8 wave32 per block
    const int wavesPerBlock = block / 32;
    int grid = 8192;                         // 65536 waves, 4 rows each (2 unrolled iters)
    const int maxBlocks = (n_rows + wavesPerBlock - 1) / wavesPerBlock;
    if (grid > maxBlocks) grid = maxBlocks;
    if (grid < 1) grid = 1;

    learned_quantizer_fwd<<<grid, block, 0, stream>>>(
        x, angles, centroids, mean_corr, running_mean, out, n_rows);
}